// DropweightLayer_5643587027453
// MI455X (gfx1250) — compile-verified
//
#include <hip/hip_runtime.h>
#include <cstdint>

// out = x * mask, mask[f] = (u[f] < drop_p[f]) ? 1 : 0, f = flat_idx % 161.
// Pure HBM-streaming kernel: ~264 MB @ 23.3 TB/s => ~11.3 us floor on MI455X.
// CDNA5 features: global_load_async_to_lds_b32 + s_wait_asynccnt for per-block
// mask staging; non-temporal (th:*_NT) B128 loads/stores for the single-pass
// 264 MB stream (exceeds 192 MB L2, zero reuse). WMMA intentionally absent:
// elementwise op, 0.125 FLOP/byte, matrix engines are off the roofline path.

typedef float f32x4 __attribute__((ext_vector_type(4)));

#define THREADS 256
#define UNROLL  8
#define F_CH    161u
#define STRIDE_MOD (1024u % F_CH)   // (THREADS*4) % 161 = 58

__global__ __launch_bounds__(THREADS)
void dropweight_mask_stream_kernel(const float* __restrict__ x,
                                   const float* __restrict__ u,
                                   const float* __restrict__ dp,
                                   float* __restrict__ out,
                                   unsigned int n)
{
    __shared__ float s_u[F_CH];
    __shared__ float s_dp[F_CH];
    __shared__ float s_mask[F_CH];

    const unsigned int tid = threadIdx.x;

    // ---- Stage u / drop_p into LDS with gfx1250 async global->LDS copies ----
    if (tid < F_CH) {
        uint32_t lds_u  = (uint32_t)(uintptr_t)(&s_u[tid]);   // low 32 bits of flat
        uint32_t lds_dp = (uint32_t)(uintptr_t)(&s_dp[tid]);  // LDS ptr = LDS offset
        uint64_t gu     = (uint64_t)(uintptr_t)(u  + tid);
        uint64_t gdp    = (uint64_t)(uintptr_t)(dp + tid);
        asm volatile("global_load_async_to_lds_b32 %0, %1, off"
                     :: "v"(lds_u), "v"(gu) : "memory");
        asm volatile("global_load_async_to_lds_b32 %0, %1, off"
                     :: "v"(lds_dp), "v"(gdp) : "memory");
    }
    asm volatile("s_wait_asynccnt 0" ::: "memory");
    __syncthreads();

    if (tid < F_CH) {
        s_mask[tid] = (s_u[tid] < s_dp[tid]) ? 1.0f : 0.0f;
    }
    __syncthreads();

    // ---- Bulk stream: 8 float4 chunks/thread, NT hints, B128 traffic ----
    const unsigned int block_elem0 = blockIdx.x * (THREADS * UNROLL * 4u);
    unsigned int e  = block_elem0 + tid * 4u;   // element index of this thread's chunk 0
    unsigned int c0 = e % F_CH;                 // ONE urem per thread; then +58 mod 161

    if (block_elem0 + THREADS * UNROLL * 4u <= n) {
        // Full block (every block except possibly the last): branch-free hot loop.
#pragma unroll
        for (int j = 0; j < UNROLL; ++j) {
            f32x4 v = __builtin_nontemporal_load((const f32x4*)(x + e));

            unsigned int c1 = c0 + 1u; if (c1 >= F_CH) c1 -= F_CH;
            unsigned int c2 = c1 + 1u; if (c2 >= F_CH) c2 -= F_CH;
            unsigned int c3 = c2 + 1u; if (c3 >= F_CH) c3 -= F_CH;

            v.x *= s_mask[c0];
            v.y *= s_mask[c1];
            v.z *= s_mask[c2];
            v.w *= s_mask[c3];

            __builtin_nontemporal_store(v, (f32x4*)(out + e));

            e  += THREADS * 4u;
            c0 += STRIDE_MOD; if (c0 >= F_CH) c0 -= F_CH;
        }
    } else {
        // Tail block: guarded, not unrolled (cold; never taken for N=128*1600*161).
#pragma unroll 1
        for (int j = 0; j < UNROLL; ++j) {
            if (e + 3u < n) {
                f32x4 v = __builtin_nontemporal_load((const f32x4*)(x + e));

                unsigned int c1 = c0 + 1u; if (c1 >= F_CH) c1 -= F_CH;
                unsigned int c2 = c1 + 1u; if (c2 >= F_CH) c2 -= F_CH;
                unsigned int c3 = c2 + 1u; if (c3 >= F_CH) c3 -= F_CH;

                v.x *= s_mask[c0];
                v.y *= s_mask[c1];
                v.z *= s_mask[c2];
                v.w *= s_mask[c3];

                __builtin_nontemporal_store(v, (f32x4*)(out + e));
            } else if (e < n) {
                unsigned int c = c0;
                for (unsigned int e2 = e; e2 < n && e2 < e + 4u; ++e2) {
                    out[e2] = x[e2] * s_mask[c];
                    c += 1u; if (c >= F_CH) c -= F_CH;
                }
            }
            e  += THREADS * 4u;
            c0 += STRIDE_MOD; if (c0 >= F_CH) c0 -= F_CH;
        }
    }
}

extern "C" void kernel_launch(void* const* d_in, const int* in_sizes, int n_in,
                              void* d_out, int out_size, void* d_ws, size_t ws_size,
                              hipStream_t stream) {
    const float* x  = (const float*)d_in[0];   // (128,1,1600,161) fp32
    const float* u  = (const float*)d_in[1];   // (161,) fp32
    const float* dp = (const float*)d_in[2];   // (161,) fp32
    float* out      = (float*)d_out;

    const unsigned int n       = (unsigned int)in_sizes[0];
    const unsigned int chunks  = (n + 3u) / 4u;                  // float4 chunks
    const unsigned int perBlk  = THREADS * UNROLL;
    const unsigned int blocks  = (chunks + perBlk - 1u) / perBlk;

    dropweight_mask_stream_kernel<<<dim3(blocks), dim3(THREADS), 0, stream>>>(
        x, u, dp, out, n);
}